// FusedQueryAtt_66589172957230
// MI455X (gfx1250) — compile-verified
//
#include <hip/hip_runtime.h>

// ---------------------------------------------------------------------------
// Masked causal prefix-sum:  out[b,i] = x[b,i] + m[b,i] * sum_{j<=i} m[b,j]*gelu(x[b,j])
// B=8, T=4096, D=1024, fp32.  Bandwidth-bound: read x once, write out once.
//
// CDNA5 mapping:
//  * TDM (tensor_load_to_lds) streams 16x64 fp32 tiles into double-buffered LDS,
//    tracked with TENSORcnt (wave 0 issues, s_wait_tensorcnt, s_barrier).
//  * Intra-tile inclusive scan = L(16x16 lower-tri ones) x G(16x16) done on the
//    matrix unit: 4 chained V_WMMA_F32_16X16X4_F32, carry folded into C.
//  * VALU only computes gelu (exact, erf-based) and the final add/mask.
// ---------------------------------------------------------------------------

typedef __attribute__((ext_vector_type(2))) float  v2f;
typedef __attribute__((ext_vector_type(8))) float  v8f;
typedef __attribute__((ext_vector_type(4))) unsigned int u32x4;
typedef __attribute__((ext_vector_type(8))) int    i32x8;
typedef __attribute__((ext_vector_type(4))) int    i32x4;

constexpr int kB = 8;
constexpr int kT = 4096;
constexpr int kD = 1024;
constexpr int TILE_T  = 16;          // WMMA M dimension (timesteps per tile)
constexpr int BLOCK_D = 64;          // channels per workgroup (16 per wave)
constexpr int WAVES   = BLOCK_D / 16;
constexpr int NT      = kT / TILE_T; // 256 tiles along T

__device__ __forceinline__ float gelu_f(float v) {
    return 0.5f * v * (1.0f + erff(v * 0.70710678118654752f));
}

// Issue one TDM 2D tile load: 16 rows x 64 fp32, row stride = D elements.
// D# layout per CDNA5 ISA ch.8 (async/tensor). gaddr = tile start (bytes),
// lds_off = LDS byte offset of destination buffer.
__device__ __forceinline__ void tdm_load_tile(unsigned lds_off,
                                              unsigned long long gaddr) {
    u32x4 g0 = {
        1u,                                          // count=1, user mode, no gather
        lds_off,                                     // lds_addr
        (unsigned)(gaddr & 0xFFFFFFFFull),           // global_addr[31:0]
        (unsigned)((gaddr >> 32) & 0x1FFFFFFull)     // global_addr[56:32]
            | 0x80000000u                            // type=2 ("image") in [127:126]
    };
    i32x8 g1 = {
        (int)(2u << 16),            // workgroup_mask=0, data_size=2 (4 bytes)
        (int)((unsigned)BLOCK_D << 16),   // tensor_dim0[15:0] in [31:16]
        (int)((unsigned)TILE_T  << 16),   // tensor_dim0[31:16]=0 | tensor_dim1[15:0]
        (int)((unsigned)BLOCK_D << 16),   // tensor_dim1[31:16]=0 | tile_dim0=64
        (int)TILE_T,                      // tile_dim1=16 | tile_dim2=0
        (int)kD,                          // tensor_dim0_stride = 1024 elements
        0,                                // stride0 hi | tensor_dim1_stride lo (unused, 2D)
        0
    };
    i32x4 z4 = {0, 0, 0, 0};              // groups 2/3 unused for 2D tile
    i32x8 z8 = {0, 0, 0, 0, 0, 0, 0, 0};  // extra group (clang-23 6-arg form)
    __builtin_amdgcn_tensor_load_to_lds(g0, g1, z4, z4, z8, 0);
}

__global__ __launch_bounds__(WAVES * 32)
void fused_causal_prefix_kernel(const float* __restrict__ x,
                                const int*   __restrict__ lengths,
                                float*       __restrict__ out) {
    __shared__ __align__(128) float tile[2][TILE_T][BLOCK_D]; // 2 x 4 KB

    const int b    = blockIdx.x;            // batch
    const int d0   = blockIdx.y * BLOCK_D;  // channel block base
    const int lane = threadIdx.x & 31;
    const int wave = threadIdx.x >> 5;
    const int n    = lane & 15;             // N column / A's M row for this lane
    const int hi   = lane >> 4;             // 0: K/rows {0,1}+4q ; 1: {2,3}+4q
    const int col  = wave * 16 + n;         // channel within block
    const int len  = lengths[b];

    // A = lower-triangular ones, split into 4 K-chunks of the 16x4 f32 A layout:
    // lanes 0-15 hold M=0..15 with K={0,1}; lanes 16-31 same M with K={2,3}.
    v2f A[4];
#pragma unroll
    for (int q = 0; q < 4; ++q) {
        A[q].x = ((4 * q + 2 * hi + 0) <= n) ? 1.0f : 0.0f;
        A[q].y = ((4 * q + 2 * hi + 1) <= n) ? 1.0f : 0.0f;
    }

    const unsigned long long xbase =
        (unsigned long long)(uintptr_t)x +
        ((unsigned long long)b * kT * kD + (unsigned long long)d0) * sizeof(float);
    // Flat LDS address: low 32 bits == LDS byte offset (ISA aperture rule).
    const unsigned lds_base  = (unsigned)(uintptr_t)&tile[0][0][0];
    const unsigned buf_bytes = TILE_T * BLOCK_D * sizeof(float);
    const unsigned long long tile_bytes =
        (unsigned long long)TILE_T * kD * sizeof(float);

    if (wave == 0) {
        tdm_load_tile(lds_base, xbase);     // prefetch tile 0 -> buffer 0
    }

    float carry = 0.0f;                     // running inclusive sum, replicated
                                            // across lanes n and n+16

    for (int i = 0; i < NT; ++i) {
        const int t0 = i * TILE_T;

        if (wave == 0) {
            if (i + 1 < NT) {
                tdm_load_tile(lds_base + (unsigned)((i + 1) & 1) * buf_bytes,
                              xbase + (unsigned long long)(i + 1) * tile_bytes);
                __builtin_amdgcn_s_wait_tensorcnt(1);   // tile i landed
            } else {
                __builtin_amdgcn_s_wait_tensorcnt(0);
            }
        }
        __syncthreads();                    // tile i visible to all waves

        float (*buf)[BLOCK_D] = tile[i & 1];

        // Accumulator C pre-loaded with carry: C[m, n] = carry[n] for all m.
        v8f acc;
#pragma unroll
        for (int v = 0; v < 8; ++v) acc[v] = carry;

        // B = G = gelu(x)*valid in the 4x16 f32 B layout; chain K=4 chunks.
#pragma unroll
        for (int q = 0; q < 4; ++q) {
            const int r  = 4 * q + 2 * hi;
            const float x0 = buf[r][col];
            const float x1 = buf[r + 1][col];
            v2f Bq;
            Bq.x = (t0 + r     < len) ? gelu_f(x0) : 0.0f;
            Bq.y = (t0 + r + 1 < len) ? gelu_f(x1) : 0.0f;
            acc = __builtin_amdgcn_wmma_f32_16x16x4_f32(
                false, A[q], false, Bq, (short)0, acc, false, false);
        }

        // D VGPR v holds rows v (lanes 0-15) and v+8 (lanes 16-31), col n.
#pragma unroll
        for (int v = 0; v < 8; ++v) {
            const int row = v + 8 * hi;
            const int t   = t0 + row;
            const float xv = buf[row][col];
            const float res = xv + ((t < len) ? acc[v] : 0.0f);
            out[((unsigned long long)b * kT + t) * kD + d0 + col] = res;
        }

        // New carry = inclusive total through row 15 (in lanes 16-31 of acc[7]);
        // broadcast so lanes n and n+16 both hold carry[n].
        carry = __shfl(acc[7], n + 16, 32);

        __syncthreads();                    // buf[i&1] free for tile i+2
    }
}

extern "C" void kernel_launch(void* const* d_in, const int* in_sizes, int n_in,
                              void* d_out, int out_size, void* d_ws, size_t ws_size,
                              hipStream_t stream) {
    const float* x       = (const float*)d_in[0];
    const int*   lengths = (const int*)d_in[1];   // jax default x64-off -> int32
    float*       out     = (float*)d_out;
    (void)in_sizes; (void)n_in; (void)out_size; (void)d_ws; (void)ws_size;

    dim3 grid(kB, kD / BLOCK_D);          // 8 x 16 = 128 workgroups
    fused_causal_prefix_kernel<<<grid, WAVES * 32, 0, stream>>>(x, lengths, out);
}